// MixedOp_6631429505500
// MI455X (gfx1250) — compile-verified
//
#include <hip/hip_runtime.h>
#include <cstdint>
#include <cstddef>

#define N_NODESC 40000
#define N_EDGESC 640000
#define DF 128
#define N_OPSC 5
#define EPSC 1e-5f

typedef float v2f __attribute__((ext_vector_type(2)));
typedef float v8f __attribute__((ext_vector_type(8)));

// Order-preserving float <-> uint mapping so float-max becomes native u32 atomicMax.
__device__ __forceinline__ unsigned enc_ord(float v) {
    unsigned b = __float_as_uint(v);
    return (b & 0x80000000u) ? ~b : (b | 0x80000000u);
}
__device__ __forceinline__ float dec_ord(unsigned u) {
    unsigned b = (u & 0x80000000u) ? (u & 0x7fffffffu) : ~u;
    return __uint_as_float(b);
}
#define ENC_NEG_INF 0x007FFFFFu  // enc_ord(-inf)

// v_rcp_f32 + one Newton-Raphson step (~0.5 ulp), avoids IEEE div expansion.
__device__ __forceinline__ float fast_rcp(float x) {
    float r = __builtin_amdgcn_rcpf(x);
    return r * (2.0f - x * r);
}

// ---------------------------------------------------------------- K1: init ws
__global__ void k_init(float* __restrict__ agg, unsigned* __restrict__ umax,
                       float* __restrict__ degstats /* deg[40000] + stats[1280] */) {
    int i = blockIdx.x * blockDim.x + threadIdx.x;   // exactly N*D threads
    agg[i]  = 0.0f;
    umax[i] = ENC_NEG_INF;
    if (i < N_NODESC + 2 * N_OPSC * DF) degstats[i] = 0.0f;
}

// ------------------------------------------------------------ K2: edge scatter
// 32 lanes per edge, 4 feats/lane (float4 gather), L2-resident atomics.
__global__ void k_scatter(const int* __restrict__ ei, const float* __restrict__ h,
                          float* __restrict__ agg, unsigned* __restrict__ umax,
                          float* __restrict__ deg) {
    int idx   = blockIdx.x * blockDim.x + threadIdx.x;   // N_EDGES*32 threads
    int e     = idx >> 5;
    int lane4 = (idx & 31) << 2;
    int s = ei[e];
    int d = ei[N_EDGESC + e];
    const float4 v = *reinterpret_cast<const float4*>(h + (size_t)s * DF + lane4);
    float* a = agg + (size_t)d * DF + lane4;
    atomicAdd(a + 0, v.x); atomicAdd(a + 1, v.y);
    atomicAdd(a + 2, v.z); atomicAdd(a + 3, v.w);
    unsigned* m = umax + (size_t)d * DF + lane4;
    atomicMax(m + 0, enc_ord(v.x)); atomicMax(m + 1, enc_ord(v.y));
    atomicMax(m + 2, enc_ord(v.z)); atomicMax(m + 3, enc_ord(v.w));
    if ((idx & 31) == 0) atomicAdd(deg + d, 1.0f);
}

// ---------------------------------------------- K3: BN stats via WMMA f32 16x16x4
// Gram trick: C += X^T X (diag = sum of squares), plus ones^T X (row 0 = sums).
// For 16x16x4 f32 the A(16x4) and B(4x16) lane layouts coincide for A=X^T,B=X,
// so the same VGPR pair feeds SRC0 and SRC1. All loads unconditional; selects
// via v_cndmask so EXEC stays all-ones around the WMMAs.
#define STAT_BLOCKS 100
__global__ void __launch_bounds__(256) k_stats(
    const float* __restrict__ h, const float* __restrict__ h_in,
    const float* __restrict__ agg, const unsigned* __restrict__ umax,
    const float* __restrict__ deg, float* __restrict__ stats /* [2][5][128] */) {
    const int lane  = threadIdx.x & 31;
    const int wave  = threadIdx.x >> 5;     // 8 waves -> 8 feature tiles of 16
    const int ftile = wave << 4;
    const int feat  = ftile + (lane & 15);
    const int half  = lane >> 4;            // 0: nodes {0,1}  1: nodes {2,3}

    const int nodesPer = N_NODESC / STAT_BLOCKS;       // 400
    const int n_begin  = blockIdx.x * nodesPer;

    v8f gram[N_OPSC], csum[N_OPSC];
#pragma unroll
    for (int b = 0; b < N_OPSC; ++b)
#pragma unroll
        for (int r = 0; r < 8; ++r) { gram[b][r] = 0.0f; csum[b][r] = 0.0f; }

    v2f ones = {1.0f, 1.0f};

    // Strength-reduced addressing: node na = n_begin + 2*half + 4*it, nb = na+1.
    size_t off = (size_t)(n_begin + (half << 1)) * DF + feat;
    int    nd  = n_begin + (half << 1);

    for (int it = 0; it < nodesPer / 4; ++it) {
        float dga = deg[nd], dgb = deg[nd + 1];
        float ra = fast_rcp(fmaxf(dga, 1.0f));
        float rb = fast_rcp(fmaxf(dgb, 1.0f));
        unsigned ua = umax[off], ub = umax[off + DF];     // unconditional loads
        float mxa = dec_ord(ua), mxb = dec_ord(ub);
        v2f xh; xh[0] = h[off];    xh[1] = h[off + DF];
        v2f xi; xi[0] = h_in[off]; xi[1] = h_in[off + DF];
        v2f xs; xs[0] = agg[off];  xs[1] = agg[off + DF];
        v2f xm; xm[0] = xs[0] * ra; xm[1] = xs[1] * rb;
        v2f xx; xx[0] = (dga > 0.5f) ? mxa : 0.0f;        // v_cndmask, no branch
                xx[1] = (dgb > 0.5f) ? mxb : 0.0f;
        v2f xb[N_OPSC] = {xh, xi, xs, xm, xx};
#pragma unroll
        for (int b = 0; b < N_OPSC; ++b) {
            gram[b] = __builtin_amdgcn_wmma_f32_16x16x4_f32(
                false, xb[b], false, xb[b], (short)0, gram[b], false, false);
            csum[b] = __builtin_amdgcn_wmma_f32_16x16x4_f32(
                false, ones,  false, xb[b], (short)0, csum[b], false, false);
        }
        off += 4 * DF;
        nd  += 4;
    }

    // Extract: sums live in row M=0 (VGPR0, lanes 0-15); sumsq on the Gram diagonal
    // (feat f: VGPR f&7, lane f (f<8) or f+16 (f>=8)). Shuffles with full EXEC.
    int srcLane = (lane & 15) + ((lane & 8) ? 16 : 0);
#pragma unroll
    for (int b = 0; b < N_OPSC; ++b) {
        float ssum = csum[b][0];
        float ssq  = 0.0f;
#pragma unroll
        for (int r = 0; r < 8; ++r) {
            float g = __shfl(gram[b][r], srcLane, 32);
            if ((lane & 7) == r) ssq = g;
        }
        if (lane < 16) {
            atomicAdd(&stats[b * DF + feat], ssum);
            atomicAdd(&stats[N_OPSC * DF + b * DF + feat], ssq);
        }
    }
}

// ------------------------------------------------- K4: fold BN into scale/shift
__global__ void k_bn(const float* __restrict__ stats, const float* __restrict__ gamma,
                     const float* __restrict__ beta, float* __restrict__ scale,
                     float* __restrict__ shift) {
    int i = blockIdx.x * blockDim.x + threadIdx.x;
    if (i >= N_OPSC * DF) return;
    const float invN = 1.0f / (float)N_NODESC;
    float mu  = stats[i] * invN;
    float var = fmaxf(stats[N_OPSC * DF + i] * invN - mu * mu, 0.0f);
    float sc  = gamma[i] * rsqrtf(var + EPSC);
    scale[i] = sc;
    shift[i] = beta[i] - mu * sc;
}

// ------------------------- K5: fused output, async-to-LDS double-buffered stream
__global__ void __launch_bounds__(256) k_out(
    const float* __restrict__ h, const float* __restrict__ h_in,
    const float* __restrict__ agg, const unsigned* __restrict__ umax,
    const float* __restrict__ deg, const float* __restrict__ scale,
    const float* __restrict__ shift, const float* __restrict__ w,
    float* __restrict__ out) {
    __shared__ __align__(16) char smem[8 * 4096];   // per-wave: 2 stages x 4 rows x 512B
    const int lane  = threadIdx.x & 31;
    const int wave  = threadIdx.x >> 5;
    const int lane4 = lane << 2;
    char* wbase = smem + wave * 4096;

    float4 sc[N_OPSC], sh[N_OPSC];
#pragma unroll
    for (int b = 0; b < N_OPSC; ++b) {
        sc[b] = *reinterpret_cast<const float4*>(scale + b * DF + lane4);
        sh[b] = *reinterpret_cast<const float4*>(shift + b * DF + lane4);
    }
    float w0 = w[0], w1 = w[1], w2 = w[2], w3 = w[3], w4 = w[4];

    const int step = gridDim.x * 8;
    int n = blockIdx.x * 8 + wave;
    if (n >= N_NODESC) return;

    auto issue = [&](int node, int stg) {
        unsigned lbase = (unsigned)(uintptr_t)(wbase + stg * 2048 + lane * 16);
        unsigned long long gh  = (unsigned long long)(uintptr_t)(h    + (size_t)node * DF + lane4);
        unsigned long long ghi = (unsigned long long)(uintptr_t)(h_in + (size_t)node * DF + lane4);
        unsigned long long ga  = (unsigned long long)(uintptr_t)(agg  + (size_t)node * DF + lane4);
        unsigned long long gu  = (unsigned long long)(uintptr_t)(umax + (size_t)node * DF + lane4);
        asm volatile("global_load_async_to_lds_b128 %0, %1, off" :: "v"(lbase +    0u), "v"(gh)  : "memory");
        asm volatile("global_load_async_to_lds_b128 %0, %1, off" :: "v"(lbase +  512u), "v"(ghi) : "memory");
        asm volatile("global_load_async_to_lds_b128 %0, %1, off" :: "v"(lbase + 1024u), "v"(ga)  : "memory");
        asm volatile("global_load_async_to_lds_b128 %0, %1, off" :: "v"(lbase + 1536u), "v"(gu)  : "memory");
    };

    auto compute = [&](int node, int stg) {
        char* p = wbase + stg * 2048;
        float4 vh  = *reinterpret_cast<float4*>(p +    0 + lane * 16);
        float4 vhi = *reinterpret_cast<float4*>(p +  512 + lane * 16);
        float4 va  = *reinterpret_cast<float4*>(p + 1024 + lane * 16);
        uint4  vu  = *reinterpret_cast<uint4*> (p + 1536 + lane * 16);
        float dg  = deg[node];
        float inv = fast_rcp(fmaxf(dg, 1.0f));
        bool  has = dg > 0.5f;
        float4 r;
#define BRANCH5(c)                                                             \
        do {                                                                   \
            float x3 = va.c * inv;                                             \
            float x4d = dec_ord(vu.c);                                         \
            float x4 = has ? x4d : 0.0f;                                       \
            r.c = w0 * fmaxf(sc[0].c * vh.c  + sh[0].c, 0.0f)                  \
                + w1 * fmaxf(sc[1].c * vhi.c + sh[1].c, 0.0f)                  \
                + w2 * fmaxf(sc[2].c * va.c  + sh[2].c, 0.0f)                  \
                + w3 * fmaxf(sc[3].c * x3    + sh[3].c, 0.0f)                  \
                + w4 * fmaxf(sc[4].c * x4    + sh[4].c, 0.0f);                 \
        } while (0)
        BRANCH5(x); BRANCH5(y); BRANCH5(z); BRANCH5(w);
#undef BRANCH5
        *reinterpret_cast<float4*>(out + (size_t)node * DF + lane4) = r;
    };

    issue(n, 0);
    int cur = n, s = 0;
    while (true) {
        int nxt = cur + step;
        bool more = (nxt < N_NODESC);
        if (more) {
            issue(nxt, s ^ 1);
            asm volatile("s_wait_asynccnt 4" ::: "memory");  // first 4 (in-order) done
        } else {
            asm volatile("s_wait_asynccnt 0" ::: "memory");
        }
        compute(cur, s);
        if (!more) break;
        cur = nxt; s ^= 1;
    }
}

// ----------------------------------------------------------------- launcher
extern "C" void kernel_launch(void* const* d_in, const int* in_sizes, int n_in,
                              void* d_out, int out_size, void* d_ws, size_t ws_size,
                              hipStream_t stream) {
    const float* weights = (const float*)d_in[0];
    const int*   ei      = (const int*)d_in[1];
    const float* h       = (const float*)d_in[2];
    const float* h_in    = (const float*)d_in[3];
    const float* gamma   = (const float*)d_in[4];
    const float* beta    = (const float*)d_in[5];
    float* out = (float*)d_out;

    // Workspace layout (~41.1 MB)
    char* ws = (char*)d_ws;
    float*    agg   = (float*)ws;                                  // 5,120,000 f32
    unsigned* umax  = (unsigned*)(ws + (size_t)N_NODESC * DF * 4); // 5,120,000 u32
    float*    deg   = (float*)(ws + 2ull * N_NODESC * DF * 4);     // 40,000 f32
    float*    stats = deg + N_NODESC;                              // 1,280 f32
    float*    scale = stats + 2 * N_OPSC * DF;                     // 640 f32
    float*    shift = scale + N_OPSC * DF;                         // 640 f32

    k_init   <<<(N_NODESC * DF) / 256, 256, 0, stream>>>(agg, umax, deg);
    k_scatter<<<(N_EDGESC * 32) / 256, 256, 0, stream>>>(ei, h, agg, umax, deg);
    k_stats  <<<STAT_BLOCKS, 256, 0, stream>>>(h, h_in, agg, umax, deg, stats);
    k_bn     <<<3, 256, 0, stream>>>(stats, gamma, beta, scale, shift);
    k_out    <<<125, 256, 0, stream>>>(h, h_in, agg, umax, deg, scale, shift, weights, out);
}